// New2_77395310674431
// MI455X (gfx1250) — compile-verified
//
#include <hip/hip_runtime.h>
#include <hip/hip_bf16.h>
#include <math.h>

// ---------------------------------------------------------------------------
// Problem constants (from reference setup_inputs)
// ---------------------------------------------------------------------------
#define B      1024
#define H      200
#define D      128
#define REGN   1000
#define STR    132            // LDS row stride (floats): 132 % 64 = 4 -> conflict-free
#define MT     13             // ceil(208/16) M tiles (rows 200..207 zero-padded)
#define NT     8              // 128/16 N tiles
#define KS     32             // 128/4  K steps

typedef float v2f __attribute__((ext_vector_type(2)));
typedef float v8f __attribute__((ext_vector_type(8)));

// LDS carve (floats)
#define OFF_HIST   0
#define SZ_HIST    (208 * STR)                 // 27456
#define OFF_K      (OFF_HIST + SZ_HIST)
#define SZ_K       (208 * STR)                 // 27456
#define OFF_W      (OFF_K + SZ_K)
#define SZ_W       (128 * STR)                 // 16896
#define OFF_TGT    (OFF_W + SZ_W)              // 128
#define OFF_Q      (OFF_TGT + 128)             // 128
#define OFF_EA     (OFF_Q + 128)               // 208
#define OFF_HDOT   (OFF_EA + 208)              // 208
#define OFF_VD     (OFF_HDOT + 208)            // 208
#define OFF_RED    (OFF_VD + 208)              // 16
#define OFF_HT     (OFF_RED + 16)              // 208 (ints)
#define SMEM_FLOATS (OFF_HT + 208)             // 72912 floats = 291,648 B  (< 320 KB)

// ---------------------------------------------------------------------------
// Kernel 1: batch-shared geo vector  gv[hj] = relu( sum_b ed[tr[b]]*ed[hr[b,hj]] )
// ---------------------------------------------------------------------------
__global__ void __launch_bounds__(256) geo_vec_kernel(
    const int* __restrict__ hreg, const int* __restrict__ treg,
    const float* __restrict__ embed_dist, const int* __restrict__ uid,
    float* __restrict__ gv) {
  const int hj  = blockIdx.x;          // 0..H-1
  const int tid = threadIdx.x;
  __shared__ float red[8];
  const float* ed = embed_dist + (size_t)uid[0] * REGN;
  float s = 0.f;
  for (int bp = tid; bp < B; bp += 256)
    s += ed[treg[bp]] * ed[hreg[bp * H + hj]];
#pragma unroll
  for (int o = 16; o > 0; o >>= 1) s += __shfl_xor(s, o);
  if ((tid & 31) == 0) red[tid >> 5] = s;
  __syncthreads();
  if (tid == 0) {
    float t = 0.f;
#pragma unroll
    for (int i = 0; i < 8; ++i) t += red[i];
    gv[hj] = fmaxf(t, 0.f);
  }
}

// ---------------------------------------------------------------------------
// Fused main kernel: one workgroup per batch row
// ---------------------------------------------------------------------------
__device__ __forceinline__ float block_sum256(float v, float* red, int tid) {
#pragma unroll
  for (int o = 16; o > 0; o >>= 1) v += __shfl_xor(v, o);
  __syncthreads();                       // protect red[] reuse across calls
  if ((tid & 31) == 0) red[tid >> 5] = v;
  __syncthreads();
  float s = red[0];
#pragma unroll
  for (int i = 1; i < 8; ++i) s += red[i];
  return s;
}

__global__ void __launch_bounds__(256) fused_rec_kernel(
    const int* __restrict__ history, const int* __restrict__ target,
    const int* __restrict__ hreg, const int* __restrict__ treg,
    const float* __restrict__ hv, const float* __restrict__ dist_mat,
    const float* __restrict__ embed_target, const float* __restrict__ embed_region,
    const float* __restrict__ Wq, const float* __restrict__ Wk,
    const float* __restrict__ Wv, const float* __restrict__ gv,
    float* __restrict__ out) {
  extern __shared__ float smem[];
  float* hist_s = smem + OFF_HIST;
  float* k_s    = smem + OFF_K;
  float* w_s    = smem + OFF_W;
  float* tgt_s  = smem + OFF_TGT;
  float* q_s    = smem + OFF_Q;
  float* ea_s   = smem + OFF_EA;
  float* hdot_s = smem + OFF_HDOT;
  float* vd_s   = smem + OFF_VD;
  float* red_s  = smem + OFF_RED;
  int*   ht_s   = (int*)(smem + OFF_HT);

  const int bi   = blockIdx.x;
  const int tid  = threadIdx.x;
  const int wid  = tid >> 5;
  const int lane = tid & 31;
  const int lm   = lane & 15;   // M/N index within tile
  const int lh   = lane >> 4;   // K-half select

  // ---------------- Phase A: stage everything into LDS --------------------
  const int tg  = target[bi];
  const int trg = treg[bi];
  for (int r = tid; r < H; r += 256) ht_s[r] = history[bi * H + r];
  if (tid < 64)       tgt_s[tid] = embed_target[(size_t)tg * 64 + tid];
  else if (tid < 128) tgt_s[tid] = embed_region[(size_t)trg * 64 + (tid - 64)];
  // Wk -> w_s (row n stride STR), float4 chunks (16B aligned: STR%4==0)
  for (int c = tid; c < 128 * 32; c += 256) {
    const int n = c >> 5, q = c & 31;
    *(float4*)(w_s + n * STR + q * 4) = *(const float4*)(Wk + n * 128 + q * 4);
  }
  // hist_emb gather: row r = [embed_target[h_idx] | embed_region[hr]]
  for (int c = tid; c < H * 32; c += 256) {
    const int r = c >> 5, q = c & 31;
    const float* src = (q < 16)
        ? (embed_target + (size_t)history[bi * H + r] * 64 + q * 4)
        : (embed_region + (size_t)hreg[bi * H + r] * 64 + (q - 16) * 4);
    *(float4*)(hist_s + r * STR + q * 4) = *(const float4*)src;
  }
  for (int i = tid; i < 8 * STR; i += 256) hist_s[H * STR + i] = 0.f;  // pad rows 200..207
  __syncthreads();

  // ---------------- Phase B: q projection + K GEMM (WMMA fp32) ------------
  if (tid < 128) {
    float s = 0.f;
    const float* wr = Wq + tid * 128;
#pragma unroll 4
    for (int i = 0; i < 128; ++i) s += tgt_s[i] * wr[i];
    q_s[tid] = s;
  }
  // k[m,n] = sum_i hist[m,i] * Wk[n,i];  A=hist (M=208,K=128), B[i,n]=Wk[n,i]
  // Tile PAIRS: two independent accumulator chains per wave (breaks the WMMA
  // RAW-on-D serial chain; shares A fragments -> 25% less LDS traffic).
  for (int p = wid; p < MT * (NT / 2); p += 8) {       // 52 pairs
    const int mt  = p >> 2;
    const int nt0 = (p & 3) * 2;
    v8f acc0 = {}, acc1 = {};
    const float* arow  = hist_s + (mt * 16 + lm) * STR + lh * 2;
    const float* brow0 = w_s + ((nt0    ) * 16 + lm) * STR + lh * 2;
    const float* brow1 = w_s + ((nt0 + 1) * 16 + lm) * STR + lh * 2;
#pragma unroll
    for (int ks = 0; ks < KS; ++ks) {
      v2f a  = *(const v2f*)(arow  + ks * 4);
      v2f b0 = *(const v2f*)(brow0 + ks * 4);
      v2f b1 = *(const v2f*)(brow1 + ks * 4);
      acc0 = __builtin_amdgcn_wmma_f32_16x16x4_f32(false, a, false, b0,
                                                   (short)0, acc0, false, false);
      acc1 = __builtin_amdgcn_wmma_f32_16x16x4_f32(false, a, false, b1,
                                                   (short)0, acc1, false, false);
    }
    float* kb0 = k_s + (mt * 16 + lh * 8) * STR + (nt0    ) * 16 + lm;
    float* kb1 = k_s + (mt * 16 + lh * 8) * STR + (nt0 + 1) * 16 + lm;
#pragma unroll
    for (int g = 0; g < 8; ++g) {        // D: VGPR g -> M=g+8*lh, N=lm
      kb0[g * STR] = acc0[g];
      kb1[g * STR] = acc1[g];
    }
  }
  __syncthreads();

  // ------- Phase C: Wv stage + reshape-scores + hist·tgt dots -------------
  for (int c = tid; c < 128 * 32; c += 256) {
    const int n = c >> 5, q = c & 31;
    *(float4*)(w_s + n * STR + q * 4) = *(const float4*)(Wv + n * 128 + q * 4);
  }
  if (tid < H) {
    // scores[hj] = sum_d q[d] * k_flat[d*H + hj]  (reshape, NOT transpose)
    float s = 0.f;
#pragma unroll 4
    for (int di = 0; di < 128; ++di) {
      const int idx = di * H + tid;
      s += q_s[di] * k_s[(idx >> 7) * STR + (idx & 127)];
    }
    s *= 0.08838834764831845f;  // 1/sqrt(128)
    ea_s[tid] = (ht_s[tid] != tg) ? expf(s) : 0.f;
    float hd = 0.f;
    const float* hrow = hist_s + tid * STR;
#pragma unroll 4
    for (int i = 0; i < 128; ++i) hd += hrow[i] * tgt_s[i];
    hdot_s[tid] = hd;
  }
  __syncthreads();

  // ---------------- Phase D: V GEMM (WMMA) fused with (v[h]·tgt) ----------
  for (int mt = wid; mt < MT; mt += 8) {
    float vdacc[8] = {0.f, 0.f, 0.f, 0.f, 0.f, 0.f, 0.f, 0.f};
    const float* arow = hist_s + (mt * 16 + lm) * STR + lh * 2;
    for (int nt = 0; nt < NT; nt += 2) {   // paired accumulators here too
      v8f acc0 = {}, acc1 = {};
      const float* brow0 = w_s + ((nt    ) * 16 + lm) * STR + lh * 2;
      const float* brow1 = w_s + ((nt + 1) * 16 + lm) * STR + lh * 2;
#pragma unroll
      for (int ks = 0; ks < KS; ++ks) {
        v2f a  = *(const v2f*)(arow  + ks * 4);
        v2f b0 = *(const v2f*)(brow0 + ks * 4);
        v2f b1 = *(const v2f*)(brow1 + ks * 4);
        acc0 = __builtin_amdgcn_wmma_f32_16x16x4_f32(false, a, false, b0,
                                                     (short)0, acc0, false, false);
        acc1 = __builtin_amdgcn_wmma_f32_16x16x4_f32(false, a, false, b1,
                                                     (short)0, acc1, false, false);
      }
      const float tv0 = tgt_s[(nt    ) * 16 + lm];
      const float tv1 = tgt_s[(nt + 1) * 16 + lm];
#pragma unroll
      for (int g = 0; g < 8; ++g) vdacc[g] += acc0[g] * tv0 + acc1[g] * tv1;
    }
#pragma unroll
    for (int g = 0; g < 8; ++g) {   // reduce over N (lanes within each 16-half)
      float x = vdacc[g];
      x += __shfl_xor(x, 1); x += __shfl_xor(x, 2);
      x += __shfl_xor(x, 4); x += __shfl_xor(x, 8);
      if (lm == 0) vd_s[mt * 16 + lh * 8 + g] = x;
    }
  }
  __syncthreads();

  // ---------------- Phase E: attention norm, geo, final contraction -------
  const float S  = block_sum256((tid < H) ? ea_s[tid] : 0.f, red_s, tid);
  const float nS = sqrtf(S);          // exp_sum ** BETA, BETA = 0.5
  float p = 0.f;
  if (tid < H) {
    const float attn = ea_s[tid] / nS;
    const float geo  = expf(-dist_mat[bi * H + tid] / (gv[tid] + 1.0f));
    p = (attn + geo) * vd_s[tid] + hv[tid] * hdot_s[tid];
  }
  const float pred = block_sum256(p, red_s, tid);
  if (tid == 0) out[bi] = 1.0f / (1.0f + expf(-pred));
}

// ---------------------------------------------------------------------------
extern "C" void kernel_launch(void* const* d_in, const int* in_sizes, int n_in,
                              void* d_out, int out_size, void* d_ws, size_t ws_size,
                              hipStream_t stream) {
  const int*   history = (const int*)d_in[0];
  const int*   target  = (const int*)d_in[1];
  const int*   hreg    = (const int*)d_in[2];
  const int*   treg    = (const int*)d_in[3];
  const float* hv      = (const float*)d_in[4];
  const float* dist    = (const float*)d_in[5];
  const int*   uid     = (const int*)d_in[6];
  const float* etgt    = (const float*)d_in[7];
  const float* ereg    = (const float*)d_in[8];
  const float* edist   = (const float*)d_in[9];
  const float* Wq      = (const float*)d_in[10];
  const float* Wk      = (const float*)d_in[11];
  const float* Wv      = (const float*)d_in[12];
  float* out = (float*)d_out;
  float* gv  = (float*)d_ws;           // H floats of scratch

  geo_vec_kernel<<<H, 256, 0, stream>>>(hreg, treg, edist, uid, gv);
  const size_t smem_bytes = (size_t)SMEM_FLOATS * sizeof(float);
  fused_rec_kernel<<<B, 256, smem_bytes, stream>>>(
      history, target, hreg, treg, hv, dist, etgt, ereg, Wq, Wk, Wv, gv, out);
}